// RandomShiftsAug_1726576854309
// MI455X (gfx1250) — compile-verified
//
#include <hip/hip_runtime.h>

// RandomShiftsAug: out[n,c,i,j] = bilinear sample of edge-padded x at
// (i + sy, j + sx) on the 144x144 padded image, reproducing the reference's
// float32 grid math exactly (coords are ~integers; bilinear handles the ulps).

#define HH 128
#define WW 128
#define CC 9
#define NN 256
#define SZ 144   // padded size
#define PADP 8

__global__ __launch_bounds__(256) void random_shifts_aug_kernel(
    const float* __restrict__ x,      // (N, C, 128, 128)
    const int*   __restrict__ shift,  // (N, 1, 1, 2) -> flat [2n]=sx, [2n+1]=sy
    float*       __restrict__ out)    // (N, C, 128, 128)
{
    __shared__ float smem[HH * WW];   // one full channel image: 64 KB

    const int nc  = blockIdx.x;       // n*9 + c
    const int n   = nc / CC;
    const int tid = threadIdx.x;

    const float* __restrict__ src = x + (size_t)nc * (HH * WW);

    // ---------------- stage channel image into LDS ----------------
    // 16384 floats = 4096 x b128; 256 threads -> 16 async copies each.
#if __has_builtin(__builtin_amdgcn_global_load_async_to_lds_b128)
    {
        typedef int v4i_t __attribute__((__vector_size__(4 * sizeof(int))));
        typedef __attribute__((address_space(1))) v4i_t* gv4p;
        typedef __attribute__((address_space(3))) v4i_t* lv4p;
        gv4p gp = (gv4p)src;    // cast away const + to AS1 vector ptr
        lv4p lp = (lv4p)smem;   // AS3 vector ptr
#pragma unroll
        for (int k = 0; k < 16; ++k) {
            const int idx = tid + k * 256;   // in 16-byte units
            __builtin_amdgcn_global_load_async_to_lds_b128(gp + idx, lp + idx,
                                                           /*offset=*/0, /*cpol=*/0);
        }
#if __has_builtin(__builtin_amdgcn_s_wait_asynccnt)
        __builtin_amdgcn_s_wait_asynccnt(0);
#else
        asm volatile("s_wait_asynccnt 0" ::: "memory");
#endif
    }
#else
    {
        const float4* g4 = (const float4*)src;
        float4* s4 = (float4*)smem;
#pragma unroll
        for (int k = 0; k < 16; ++k) {
            const int idx = tid + k * 256;
            s4[idx] = g4[idx];
        }
    }
#endif
    __syncthreads();

    // ---------------- per-image shift (uniform across block) ----------------
    const int sx = shift[2 * n + 0];
    const int sy = shift[2 * n + 1];

    const float eps   = 1.0f / (float)SZ;
    const float start = -1.0f + eps;
    const float stop  =  1.0f - eps;
    const float step  = (stop - start) / (float)(SZ - 1);
    const float shx   = (float)sx * (2.0f / (float)SZ);
    const float shy   = (float)sy * (2.0f / (float)SZ);

    // ------------- x-coordinate math: fixed per thread (column j) -------------
    const int j = tid & (WW - 1);
    const float gx  = (start + (float)j * step) + shx;
    const float ix  = ((gx + 1.0f) * (float)SZ - 1.0f) * 0.5f;
    const float ix0f = floorf(ix);
    const float wx   = ix - ix0f;
    const int   ix0  = (int)ix0f;

    const int xpa = ix0;                           // padded-space columns
    const int xpb = ix0 + 1;
    const float vxa = (xpa >= 0 && xpa < SZ) ? 1.0f : 0.0f;
    const float vxb = (xpb >= 0 && xpb < SZ) ? 1.0f : 0.0f;
    const int xa = min(max(xpa - PADP, 0), WW - 1); // edge-clamped source cols
    const int xb = min(max(xpb - PADP, 0), WW - 1);

    float* __restrict__ dst = out + (size_t)nc * (HH * WW);
    const int ihalf = tid >> 7;                    // two rows per iteration

#pragma unroll 4
    for (int t = 0; t < HH / 2; ++t) {
        const int i = 2 * t + ihalf;

        const float gy  = (start + (float)i * step) + shy;
        const float iy  = ((gy + 1.0f) * (float)SZ - 1.0f) * 0.5f;
        const float iy0f = floorf(iy);
        const float wy   = iy - iy0f;
        const int   iy0  = (int)iy0f;

        const int ypa = iy0;
        const int ypb = iy0 + 1;
        const float vya = (ypa >= 0 && ypa < SZ) ? 1.0f : 0.0f;
        const float vyb = (ypb >= 0 && ypb < SZ) ? 1.0f : 0.0f;
        const int ya = min(max(ypa - PADP, 0), HH - 1);
        const int yb = min(max(ypb - PADP, 0), HH - 1);

        const float v00 = smem[ya * WW + xa] * (vya * vxa);
        const float v01 = smem[ya * WW + xb] * (vya * vxb);
        const float v10 = smem[yb * WW + xa] * (vyb * vxa);
        const float v11 = smem[yb * WW + xb] * (vyb * vxb);

        const float r = v00 * (1.0f - wx) * (1.0f - wy)
                      + v01 * wx          * (1.0f - wy)
                      + v10 * (1.0f - wx) * wy
                      + v11 * wx          * wy;

        dst[i * WW + j] = r;
    }
}

extern "C" void kernel_launch(void* const* d_in, const int* in_sizes, int n_in,
                              void* d_out, int out_size, void* d_ws, size_t ws_size,
                              hipStream_t stream) {
    (void)in_sizes; (void)n_in; (void)out_size; (void)d_ws; (void)ws_size;
    const float* x     = (const float*)d_in[0];
    const int*   shift = (const int*)d_in[1];
    float*       out   = (float*)d_out;

    dim3 grid(NN * CC);   // 2304 blocks: one (n, c) channel image each
    dim3 block(256);      // 8 wave32s
    hipLaunchKernelGGL(random_shifts_aug_kernel, grid, block, 0, stream,
                       x, shift, out);
}